// QuantumEmbedding_4415226380342
// MI455X (gfx1250) — compile-verified
//
#include <hip/hip_runtime.h>

// 12-qubit state-vector simulator, one workgroup per batch element.
// State (4096 complex f32 = 32KB) lives in LDS. Single-qubit gates are fused
// per adjacent wire pair into a 4x4 complex (= 8x8 real) matrix, duplicated
// into both row-halves of the WMMA A operand, and applied as two chained
// V_WMMA_F32_16X16X4_F32 ops per 16 amplitude groups. All 8 rounds of B
// operands are burst-preloaded (in-order DS) so ds latency overlaps WMMAs,
// and every lane does branch-free loads AND stores (no exec juggling).

typedef __attribute__((ext_vector_type(2))) float v2f;
typedef __attribute__((ext_vector_type(8))) float v8f;

#define NQ      12
#define DIM     4096
#define THREADS 256
#define NWAVES  8
#define NLAYERS 6
#define NPAIRS  6             // wire pairs (0,1)(2,3)...(10,11)
#define NROUNDS 8             // 1024 groups / (8 waves * 16 cols)

__device__ __forceinline__ float2 cmul(float2 a, float2 b) {
    return make_float2(a.x * b.x - a.y * b.y, a.x * b.y + a.y * b.x);
}

// U = RZ(tz) * RY(ty) * RX(tx)   (2x2 complex, row-major U[r*2+c])
__device__ __forceinline__ void fused_rot(float tx, float ty, float tz, float2* U) {
    const float cx = cosf(0.5f * tx), sx = sinf(0.5f * tx);
    const float cy = cosf(0.5f * ty), sy = sinf(0.5f * ty);
    const float cz = cosf(0.5f * tz), sz = sinf(0.5f * tz);
    const float2 A00 = make_float2( cy * cx,  sy * sx);   // RY*RX
    const float2 A01 = make_float2(-sy * cx, -cy * sx);
    const float2 A10 = make_float2( sy * cx, -cy * sx);
    const float2 A11 = make_float2( cy * cx, -sy * sx);
    const float2 e  = make_float2(cz, -sz);               // exp(-i tz/2)
    const float2 ec = make_float2(cz,  sz);
    U[0] = cmul(e, A00);  U[1] = cmul(e, A01);
    U[2] = cmul(ec, A10); U[3] = cmul(ec, A11);
}

// U = RZ(0.5 f) * RY(f)
__device__ __forceinline__ void embed_rot(float f, float2* U) {
    const float cy = cosf(0.5f * f),  sy = sinf(0.5f * f);
    const float cz = cosf(0.25f * f), sz = sinf(0.25f * f);
    U[0] = make_float2( cz * cy, -sz * cy);
    U[1] = make_float2(-cz * sy,  sz * sy);
    U[2] = make_float2( cz * sy,  sz * sy);
    U[3] = make_float2( cz * cy,  sz * cy);
}

// M(4x4 complex) = Ub (x) Ua : M[b'a'][ba] = Ub[b'b] * Ua[a'a]
__device__ __forceinline__ void kron22(const float2* Ub, const float2* Ua, float2* M) {
    #pragma unroll
    for (int o = 0; o < 4; ++o)
        #pragma unroll
        for (int j = 0; j < 4; ++j)
            M[o * 4 + j] = cmul(Ub[(o >> 1) * 2 + (j >> 1)], Ua[(o & 1) * 2 + (j & 1)]);
}

// Apply 4x4 complex pair-gate M to wires (w, w+1) of state st.
// Group g has 4 amplitudes at base + j*(1<<w), j = (bit_{w+1}<<1)|bit_w.
// Real 8x8 matrix R rows = (re00',im00',re01',im01',re10',im10',re11',im11');
// A rows 0-7 = R, rows 8-15 = R again so both lane halves receive D rows.
__device__ __forceinline__ void apply_pair_wmma(float2* st, const float2* M,
                                                int w, int lane, int wave) {
    const int col = lane & 15;     // A row (mod 8) / B-D column N
    const int hiK = lane >> 4;     // 0: K=0,1   1: K=2,3 within each tile
    const int o   = (col >> 1) & 3;              // complex output index (rows dup'd)
    const float2 m1 = M[o * 4 + hiK];            // input j = hiK     (A1)
    const float2 m2 = M[o * 4 + hiK + 2];        // input j = hiK + 2 (A2)
    float a10, a11, a20, a21;
    if ((col & 1) == 0) { a10 = m1.x; a11 = -m1.y; a20 = m2.x; a21 = -m2.y; }
    else                { a10 = m1.y; a11 =  m1.x; a20 = m2.y; a21 =  m2.x; }
    v2f A1; A1[0] = a10; A1[1] = a11;
    v2f A2; A2[0] = a20; A2[1] = a21;

    const int s       = 1 << w;
    const int lowmask = s - 1;
    const int off1    = hiK ? s : 0;             // B1 amp: j = 0 / 1
    const int off2    = off1 + 2 * s;            // B2 amp: j = 2 / 3
    const int jst     = hiK ? 2 : 0;             // this lane stores j = jst, jst+1

    // Burst-preload every round's operands (DS is in-order per wave).
    int    bases[NROUNDS];
    float2 p1[NROUNDS], p2[NROUNDS];
    #pragma unroll
    for (int r = 0; r < NROUNDS; ++r) {
        const int g = r * (NWAVES * 16) + wave * 16 + col;
        bases[r] = ((g >> w) << (w + 2)) | (g & lowmask);
        p1[r] = st[bases[r] + off1];
        p2[r] = st[bases[r] + off2];
    }
    #pragma unroll
    for (int r = 0; r < NROUNDS; ++r) {
        v2f B1; B1[0] = p1[r].x; B1[1] = p1[r].y;
        v2f B2; B2[0] = p2[r].x; B2[1] = p2[r].y;
        v8f C = {0.f, 0.f, 0.f, 0.f, 0.f, 0.f, 0.f, 0.f};
        C = __builtin_amdgcn_wmma_f32_16x16x4_f32(false, A1, false, B1,
                                                  (short)0, C, false, false);
        C = __builtin_amdgcn_wmma_f32_16x16x4_f32(false, A2, false, B2,
                                                  (short)0, C, false, false);
        // Lanes 0-15: C[0..3] = amps j=0,1.  Lanes 16-31: C[4..7] = amps j=2,3.
        const float2 oa = hiK ? make_float2(C[4], C[5]) : make_float2(C[0], C[1]);
        const float2 ob = hiK ? make_float2(C[6], C[7]) : make_float2(C[2], C[3]);
        st[bases[r] + jst * s]       = oa;
        st[bases[r] + (jst + 1) * s] = ob;
    }
}

__global__ __launch_bounds__(THREADS)
void qsim_kernel(const float* __restrict__ fv, const float* __restrict__ theta,
                 float* __restrict__ out) {
    __shared__ float2 st[DIM];
    __shared__ float2 embM[NPAIRS][16];
    __shared__ float2 layM[NLAYERS * NPAIRS][16];
    __shared__ float  wpart[NWAVES][NQ];

    const int b    = blockIdx.x;
    const int tid  = threadIdx.x;
    const int lane = tid & 31;
    const int wave = tid >> 5;

    // |0...0>
    for (int i = tid; i < DIM; i += THREADS) st[i] = make_float2(0.f, 0.f);
    if (tid == 0) st[0] = make_float2(1.f, 0.f);

    // Precompute all pair-gate matrices in parallel (42 independent tasks).
    if (tid < NPAIRS + NLAYERS * NPAIRS) {
        float2 Ua[4], Ub[4];
        if (tid < NPAIRS) {                       // embedding (batch-dependent)
            embed_rot(fv[b * NQ + 2 * tid],     Ua);
            embed_rot(fv[b * NQ + 2 * tid + 1], Ub);
            kron22(Ub, Ua, &embM[tid][0]);
        } else {                                  // layers (batch-independent)
            const int t = tid - NPAIRS, l = t / NPAIRS, p = t % NPAIRS;
            const float* ta = theta + (l * NQ + 2 * p) * 3;
            fused_rot(ta[0], ta[1], ta[2], Ua);
            fused_rot(ta[3], ta[4], ta[5], Ub);
            kron22(Ub, Ua, &layM[t][0]);
        }
    }
    __syncthreads();

    // ---- Embedding: 6 pair-gates ----
    for (int p = 0; p < NPAIRS; ++p) {
        apply_pair_wmma(st, &embM[p][0], 2 * p, lane, wave);
        __syncthreads();
    }

    // ---- Layers: 6 pair-gates + CNOT ring ----
    for (int l = 0; l < NLAYERS; ++l) {
        for (int p = 0; p < NPAIRS; ++p) {
            apply_pair_wmma(st, &layM[l * NPAIRS + p][0], 2 * p, lane, wave);
            __syncthreads();
        }
        for (int k = 0; k < NQ; ++k) {
            const int c = (k < NQ - 1) ? k : NQ - 1;
            const int t = (k < NQ - 1) ? k + 1 : 0;
            const int lo = c < t ? c : t;
            const int hi = c < t ? t : c;
            const int midbits = hi - lo - 1;
            for (int p = tid; p < DIM / 4; p += THREADS) {
                const int i    = p & ((1 << lo) - 1);
                const int rest = p >> lo;
                const int mid  = rest & ((1 << midbits) - 1);
                const int top  = rest >> midbits;
                const int idx  = i | (mid << (lo + 1)) | (top << (hi + 1)) | (1 << c);
                const int jdx  = idx | (1 << t);
                float2 tmp = st[idx];
                st[idx] = st[jdx];
                st[jdx] = tmp;
            }
            __syncthreads();
        }
    }

    // ---- <Z_w> : deterministic wave32 shuffle + LDS reduction ----
    float acc[NQ];
    #pragma unroll
    for (int w = 0; w < NQ; ++w) acc[w] = 0.f;
    for (int i = tid; i < DIM; i += THREADS) {
        const float2 a  = st[i];
        const float  pz = a.x * a.x + a.y * a.y;
        #pragma unroll
        for (int w = 0; w < NQ; ++w)
            acc[w] += ((i >> w) & 1) ? -pz : pz;
    }
    #pragma unroll
    for (int off = 16; off > 0; off >>= 1) {
        #pragma unroll
        for (int w = 0; w < NQ; ++w)
            acc[w] += __shfl_xor(acc[w], off, 32);
    }
    if (lane == 0) {
        #pragma unroll
        for (int w = 0; w < NQ; ++w) wpart[wave][w] = acc[w];
    }
    __syncthreads();
    if (tid < NQ) {
        float s = 0.f;
        for (int q = 0; q < NWAVES; ++q) s += wpart[q][tid];
        out[b * NQ + tid] = s;
    }
}

extern "C" void kernel_launch(void* const* d_in, const int* in_sizes, int n_in,
                              void* d_out, int out_size, void* d_ws, size_t ws_size,
                              hipStream_t stream) {
    const float* fv    = (const float*)d_in[0];   // (BATCH, 12) float32
    const float* theta = (const float*)d_in[1];   // (6, 12, 3)  float32
    float*       out   = (float*)d_out;           // (BATCH, 12) float32
    const int batch = in_sizes[0] / NQ;
    qsim_kernel<<<batch, THREADS, 0, stream>>>(fv, theta, out);
}